// H_MAx_C_MaxAtt_47425028882581
// MI455X (gfx1250) — compile-verified
//
#include <hip/hip_runtime.h>

// ---------------------------------------------------------------------------
// MI455X (gfx1250, wave32) implementation of the coordinate-attention module.
// Bandwidth-bound: 2 streaming passes over x (256 MiB) + 1 write of out
// (~768 MiB total -> ~33 us floor at 23.3 TB/s).
// f32 WMMA (V_WMMA_F32_16X16X4_F32) used for the fc (16x256x256) and a_h
// (256x8x128 per batch) GEMMs -- exact-precision matrix path on CDNA5.
// ---------------------------------------------------------------------------

typedef __attribute__((ext_vector_type(2))) float v2f;
typedef __attribute__((ext_vector_type(4))) float v4f;
typedef __attribute__((ext_vector_type(8))) float v8f;

#define N_  16
#define C_  256
#define H_  128
#define W_  128
#define MIP 8
#define EPSF 1e-5f

// sigmoid via raw v_exp_f32 + v_rcp_f32 (no IEEE divide expansion)
__device__ __forceinline__ float sigmoidf_(float v) {
    return __builtin_amdgcn_rcpf(1.0f + __expf(-v));
}

// ---------------------------------------------------------------------------
// Kernel 1: per-row (n,c,h) pool over W: rowmean, rowmax, x_h = mean+max.
// One wave per row of 128 floats; each lane loads one float4 (b128 load).
// ---------------------------------------------------------------------------
__global__ void pool_w_kernel(const float* __restrict__ x,
                              float* __restrict__ rowmean,
                              float* __restrict__ rowmax,
                              float* __restrict__ xh) {
    const int wave = (blockIdx.x * blockDim.x + threadIdx.x) >> 5;  // row id
    const int lane = threadIdx.x & 31;
    const v4f v = reinterpret_cast<const v4f*>(x)[(size_t)wave * 32 + lane];
    float s = v.x + v.y + v.z + v.w;
    float m = fmaxf(fmaxf(v.x, v.y), fmaxf(v.z, v.w));
#pragma unroll
    for (int off = 16; off; off >>= 1) {
        s += __shfl_xor(s, off, 32);
        m = fmaxf(m, __shfl_xor(m, off, 32));
    }
    if (lane == 0) {
        const float mean = s * (1.0f / (float)W_);
        rowmean[wave] = mean;
        rowmax[wave]  = m;
        xh[wave]      = mean + m;
    }
}

// ---------------------------------------------------------------------------
// Kernel 2: reduce over H: g[n,c] = mean_h(rowmean) + max_h(rowmax).
// One wave per (n,c); 32 lanes x float4 = 128 h values.
// ---------------------------------------------------------------------------
__global__ void g_reduce_kernel(const float* __restrict__ rowmean,
                                const float* __restrict__ rowmax,
                                float* __restrict__ g) {
    const int wave = (blockIdx.x * blockDim.x + threadIdx.x) >> 5;  // n*C + c
    const int lane = threadIdx.x & 31;
    const v4f sm = reinterpret_cast<const v4f*>(rowmean)[(size_t)wave * 32 + lane];
    const v4f mm = reinterpret_cast<const v4f*>(rowmax)[(size_t)wave * 32 + lane];
    float s = sm.x + sm.y + sm.z + sm.w;
    float m = fmaxf(fmaxf(mm.x, mm.y), fmaxf(mm.z, mm.w));
#pragma unroll
    for (int off = 16; off; off >>= 1) {
        s += __shfl_xor(s, off, 32);
        m = fmaxf(m, __shfl_xor(m, off, 32));
    }
    if (lane == 0) g[wave] = s * (1.0f / (float)H_) + m;
}

// ---------------------------------------------------------------------------
// Kernel 3: y = hswish(BN(conv1_w @ x_h + b)).  [16,8,128] outputs, 256 MACs
// each -- scalar VALU is the right tool (M=8 doesn't fill a WMMA tile).
// ---------------------------------------------------------------------------
__global__ void conv1_bn_hswish_kernel(const float* __restrict__ xh,
                                       const float* __restrict__ w,
                                       const float* __restrict__ b,
                                       const float* __restrict__ gamma,
                                       const float* __restrict__ beta,
                                       const float* __restrict__ mean,
                                       const float* __restrict__ var,
                                       float* __restrict__ yact) {
    const int t = blockIdx.x * blockDim.x + threadIdx.x;   // [0, N*MIP*H)
    const int h = t & (H_ - 1);
    const int m = (t >> 7) & (MIP - 1);
    const int n = t >> 10;
    const float* wr = w + m * C_;
    const float* xr = xh + (size_t)n * C_ * H_ + h;
    float s = 0.0f;
#pragma unroll 8
    for (int c = 0; c < C_; ++c) s = fmaf(wr[c], xr[(size_t)c * H_], s);
    s += b[m];
    const float scale = gamma[m] * rsqrtf(var[m] + EPSF);
    s = (s - mean[m]) * scale + beta[m];
    yact[t] = s * fminf(fmaxf(s + 3.0f, 0.0f), 6.0f) * (1.0f / 6.0f);
}

// ---------------------------------------------------------------------------
// Kernel 4: a_h = sigmoid(conv_h_w[256,8] @ yact[n][8,128] + b) via f32 WMMA.
// Per wave: one 16x16 output tile, K=8 as two 16x16x4 steps.
// A frag (16x4 f32, ISA 7.12.2): lanes 0-15 hold K={k0,k0+1}, lanes 16-31
// hold K={k0+2,k0+3}, M = lane&15.  B frag (4x16): VGPR v holds row K=k0+v
// (lanes 0-15) and K=k0+v+2 (lanes 16-31), N = lane&15.
// ---------------------------------------------------------------------------
__global__ void ah_wmma_kernel(const float* __restrict__ whw,  // [C, MIP]
                               const float* __restrict__ whb,  // [C]
                               const float* __restrict__ yact, // [N, MIP, H]
                               float* __restrict__ ah) {       // [N, C, H]
    const int gw   = (blockIdx.x * blockDim.x + threadIdx.x) >> 5;  // 0..2047
    const int lane = threadIdx.x & 31;
    const int ht = gw & 7;          // h tile (8)
    const int ct = (gw >> 3) & 15;  // c tile (16)
    const int n  = gw >> 7;         // batch (16)
    const int l15 = lane & 15;
    const int hi  = lane >> 4;      // 0 | 1
    const int h   = ht * 16 + l15;

    v8f acc = (v8f)0.0f;
#pragma unroll
    for (int k0 = 0; k0 < MIP; k0 += 4) {
        const int k = k0 + (hi << 1);
        v2f a, bfr;
        a[0]   = whw[(ct * 16 + l15) * MIP + k];
        a[1]   = whw[(ct * 16 + l15) * MIP + k + 1];
        bfr[0] = yact[((size_t)n * MIP + k) * H_ + h];
        bfr[1] = yact[((size_t)n * MIP + k + 1) * H_ + h];
        acc = __builtin_amdgcn_wmma_f32_16x16x4_f32(false, a, false, bfr,
                                                    (short)0, acc, false, false);
    }
    // D layout: VGPR r = rows M=r (lanes 0-15) and M=r+8 (lanes 16-31).
#pragma unroll
    for (int r = 0; r < 8; ++r) {
        const int c = ct * 16 + r + (hi << 3);
        const float v = acc[r] + whb[c];
        ah[((size_t)n * C_ + c) * H_ + h] = sigmoidf_(v);
    }
}

// ---------------------------------------------------------------------------
// Kernel 5: ca = sigmoid(g @ fc_w^T + fc_b).  M=16 exactly = one WMMA row.
// One block, 16 waves; wave w owns output column tile j in [16w, 16w+16).
// K = 256 stepped by 4 with V_WMMA_F32_16X16X4_F32.
// B[k][j] = fc_w[j*256 + k]  (g @ fc_w.T).
// ---------------------------------------------------------------------------
__global__ void fc_wmma_kernel(const float* __restrict__ g,    // [N, C]
                               const float* __restrict__ fw,   // [C, C]
                               const float* __restrict__ fb,   // [C]
                               float* __restrict__ ca) {       // [N, C]
    const int wave = threadIdx.x >> 5;  // 0..15 -> j tile
    const int lane = threadIdx.x & 31;
    const int l15 = lane & 15;
    const int hi  = lane >> 4;
    const int j   = wave * 16 + l15;

    v8f acc = (v8f)0.0f;
#pragma unroll 4
    for (int k0 = 0; k0 < C_; k0 += 4) {
        const int k = k0 + (hi << 1);
        v2f a, bfr;
        a[0]   = g[l15 * C_ + k];
        a[1]   = g[l15 * C_ + k + 1];
        bfr[0] = fw[j * C_ + k];
        bfr[1] = fw[j * C_ + k + 1];
        acc = __builtin_amdgcn_wmma_f32_16x16x4_f32(false, a, false, bfr,
                                                    (short)0, acc, false, false);
    }
#pragma unroll
    for (int r = 0; r < 8; ++r) {
        const int m = r + (hi << 3);            // batch row
        const float v = acc[r] + fb[j];
        ca[m * C_ + j] = sigmoidf_(v);
    }
}

// ---------------------------------------------------------------------------
// Kernel 6: out = x * a_h[n,c,h] * ca[n,c].  Pure streaming: b128 loads,
// nontemporal b128 stores (don't evict x from the 192MB L2).
// ---------------------------------------------------------------------------
__global__ void apply_kernel(const float* __restrict__ x,
                             const float* __restrict__ ah,
                             const float* __restrict__ ca,
                             float* __restrict__ out) {
    const size_t i4 = (size_t)blockIdx.x * blockDim.x + threadIdx.x; // float4 idx
    const int row = (int)(i4 >> 5);        // (n*C + c)*H + h
    const int nc  = row >> 7;              // n*C + c
    const float s = ah[row] * ca[nc];
    v4f v = reinterpret_cast<const v4f*>(x)[i4];
    v.x *= s; v.y *= s; v.z *= s; v.w *= s;
    __builtin_nontemporal_store(v, reinterpret_cast<v4f*>(out) + i4);
}

// ---------------------------------------------------------------------------
extern "C" void kernel_launch(void* const* d_in, const int* in_sizes, int n_in,
                              void* d_out, int out_size, void* d_ws, size_t ws_size,
                              hipStream_t stream) {
    const float* x       = (const float*)d_in[0];
    const float* conv1_w = (const float*)d_in[1];
    const float* conv1_b = (const float*)d_in[2];
    const float* bn_g    = (const float*)d_in[3];
    const float* bn_b    = (const float*)d_in[4];
    const float* bn_m    = (const float*)d_in[5];
    const float* bn_v    = (const float*)d_in[6];
    const float* convh_w = (const float*)d_in[7];
    const float* convh_b = (const float*)d_in[8];
    const float* fc_w    = (const float*)d_in[9];
    const float* fc_b    = (const float*)d_in[10];
    float* out = (float*)d_out;

    // Workspace layout (floats), ~8.5 MB total
    float* ws = (float*)d_ws;
    const size_t R = (size_t)N_ * C_ * H_;   // 524288 rows
    float* rowmean = ws;                     // R
    float* rowmax  = rowmean + R;            // R
    float* xh      = rowmax + R;             // R
    float* g       = xh + R;                 // N*C = 4096
    float* yact    = g + (size_t)N_ * C_;    // N*MIP*H = 16384
    float* ah      = yact + (size_t)N_ * MIP * H_;  // R
    float* ca      = ah + R;                 // N*C = 4096

    // 1) width pools: one wave per (n,c,h) row
    pool_w_kernel<<<(unsigned)(R / 8), 256, 0, stream>>>(x, rowmean, rowmax, xh);

    // 2) global pools: one wave per (n,c)
    g_reduce_kernel<<<(N_ * C_) / 8, 256, 0, stream>>>(rowmean, rowmax, g);

    // 3) conv1 + BN + h-swish: one thread per (n,m,h)
    conv1_bn_hswish_kernel<<<(N_ * MIP * H_) / 256, 256, 0, stream>>>(
        xh, conv1_w, conv1_b, bn_g, bn_b, bn_m, bn_v, yact);

    // 4) fc + sigmoid via f32 WMMA: one block, 16 waves
    fc_wmma_kernel<<<1, 512, 0, stream>>>(g, fc_w, fc_b, ca);

    // 5) a_h via f32 WMMA: 2048 waves (16 n x 16 c-tiles x 8 h-tiles)
    ah_wmma_kernel<<<(N_ * 16 * 8) / 8, 256, 0, stream>>>(convh_w, convh_b, yact, ah);

    // 6) combine: one thread per float4 of x/out
    const size_t total4 = (size_t)N_ * C_ * H_ * W_ / 4;
    apply_kernel<<<(unsigned)(total4 / 256), 256, 0, stream>>>(x, ah, ca, out);
}